// HOGLayer_79731772883056
// MI455X (gfx1250) — compile-verified
//
#include <hip/hip_runtime.h>
#include <math.h>
#include <stdint.h>

#define ORI     9
#define CELLSZ  8
#define IMGH    512
#define IMGW    512
#define NIMG    64
#define NCELL   64            // cells per side
#define TCY     16            // cells per tile, y
#define TCX     16            // cells per tile, x
#define TPY     (TCY * CELLSZ)   // 128 pixels
#define TPX     (TCX * CELLSZ)   // 128 pixels
#define LROWS   (TPY + 2)        // 130 (halo)
#define LSTRIDE 132              // floats per LDS row (528B, 16B multiple)
#define LOFF    3                // halo col c stored at slot c+LOFF -> interior c=1 is 16B aligned
#define NCHUNK  34               // per row: 1 x b32 | 32 x b128 | 1 x b32

#ifndef __has_builtin
#define __has_builtin(x) 0
#endif

#if defined(__HIP_DEVICE_COMPILE__) && \
    __has_builtin(__builtin_amdgcn_global_load_async_to_lds_b128) && \
    __has_builtin(__builtin_amdgcn_global_load_async_to_lds_b32)
#define ASYNC_BUILTIN 1
#else
#define ASYNC_BUILTIN 0
#endif

typedef int v4i __attribute__((vector_size(16)));

__device__ __forceinline__ void async_copy_b128(const float* g, float* l) {
#if ASYNC_BUILTIN
  typedef __attribute__((address_space(1))) v4i* gp128;
  typedef __attribute__((address_space(3))) v4i* lp128;
  // low 32 bits of a flat LDS address are the LDS byte offset (ISA 10.2)
  __builtin_amdgcn_global_load_async_to_lds_b128(
      (gp128)(uintptr_t)g, (lp128)(uint32_t)(uintptr_t)l, 0, 0);
#else
  asm volatile("global_load_async_to_lds_b128 %0, %1, off"
               :: "v"((uint32_t)(uintptr_t)l), "v"((uint64_t)(uintptr_t)g)
               : "memory");
#endif
}

__device__ __forceinline__ void async_copy_b32(const float* g, float* l) {
#if ASYNC_BUILTIN
  typedef __attribute__((address_space(1))) int* gp32;
  typedef __attribute__((address_space(3))) int* lp32;
  __builtin_amdgcn_global_load_async_to_lds_b32(
      (gp32)(uintptr_t)g, (lp32)(uint32_t)(uintptr_t)l, 0, 0);
#else
  asm volatile("global_load_async_to_lds_b32 %0, %1, off"
               :: "v"((uint32_t)(uintptr_t)l), "v"((uint64_t)(uintptr_t)g)
               : "memory");
#endif
}

__device__ __forceinline__ void wait_async0() {
#if defined(__HIP_DEVICE_COMPILE__) && __has_builtin(__builtin_amdgcn_s_wait_asynccnt)
  __builtin_amdgcn_s_wait_asynccnt(0);
#else
  asm volatile("s_wait_asynccnt 0" ::: "memory");
#endif
}

__global__ void __launch_bounds__(256)
hog_fused_kernel(const float* __restrict__ x, float* __restrict__ out) {
  extern __shared__ float tile[];   // LROWS * LSTRIDE floats (~68.6 KB)

  const int b   = blockIdx.x;
  const int n   = b >> 4;                 // 16 tiles per image
  const int t   = b & 15;
  const int tcy = (t >> 2) * TCY;         // cell-coordinate base of tile
  const int tcx = (t & 3)  * TCX;
  const int y0  = tcy * CELLSZ;           // pixel base of tile
  const int x0  = tcx * CELLSZ;

  const float* img = x + (size_t)n * IMGH * IMGW;

  // ---------------- stage 130x130 halo tile into LDS (async data mover) ----------------
  for (int id = threadIdx.x; id < LROWS * NCHUNK; id += 256) {
    const int r  = id / NCHUNK;
    const int k  = id - r * NCHUNK;
    const int gy = y0 - 1 + r;
    const bool rowok = (gy >= 0) && (gy < IMGH);
    float* lrow = tile + r * LSTRIDE;

    if (k == 0) {                                   // left halo column (c = 0)
      const int gx = x0 - 1;
      if (rowok && gx >= 0)
        async_copy_b32(img + (size_t)gy * IMGW + gx, lrow + LOFF + 0);
      else
        lrow[LOFF + 0] = 0.0f;
    } else if (k == NCHUNK - 1) {                   // right halo column (c = 129)
      const int gx = x0 + TPX;
      if (rowok && gx < IMGW)
        async_copy_b32(img + (size_t)gy * IMGW + gx, lrow + LOFF + TPX + 1);
      else
        lrow[LOFF + TPX + 1] = 0.0f;
    } else {                                        // interior: 16B-aligned b128 chunks
      const int c  = 1 + (k - 1) * 4;               // c in {1,5,...,125}, covers 1..128
      const int gx = x0 + (c - 1);                  // always in [0, IMGW) horizontally
      if (rowok) {
        async_copy_b128(img + (size_t)gy * IMGW + gx, lrow + LOFF + c);
      } else {
        float4 z = {0.f, 0.f, 0.f, 0.f};
        *reinterpret_cast<float4*>(lrow + LOFF + c) = z;
      }
    }
  }
  wait_async0();
  __syncthreads();

  // ---------------- one 8x8 cell per thread, separable Sobel from LDS ----------------
  const int lcx = threadIdx.x & 15;
  const int lcy = threadIdx.x >> 4;
  const int py0 = lcy * CELLSZ;   // tile-local (halo-coord) pixel window rows py0..py0+9
  const int px0 = lcx * CELLSZ;   // cols px0..px0+9

  float acc[ORI];
#pragma unroll
  for (int o = 0; o < ORI; ++o) acc[o] = 0.0f;

  float r0[10], r1[10], r2[10];
#pragma unroll
  for (int j = 0; j < 10; ++j) r0[j] = tile[(py0 + 0) * LSTRIDE + px0 + j + LOFF];
#pragma unroll
  for (int j = 0; j < 10; ++j) r1[j] = tile[(py0 + 1) * LSTRIDE + px0 + j + LOFF];

  for (int py = 0; py < CELLSZ; ++py) {
#pragma unroll
    for (int j = 0; j < 10; ++j) r2[j] = tile[(py0 + py + 2) * LSTRIDE + px0 + j + LOFF];

    float v[10], d[10];
#pragma unroll
    for (int j = 0; j < 10; ++j) {
      v[j] = r0[j] + 2.0f * r1[j] + r2[j];   // vertical (1,2,1) column sum
      d[j] = r0[j] - r2[j];                  // top - bottom row diff
    }

#pragma unroll
    for (int px = 0; px < CELLSZ; ++px) {
      const float gxv = v[px] - v[px + 2];                          // Sobel x (cross-corr)
      const float gyv = d[px] + 2.0f * d[px + 1] + d[px + 2];       // Sobel y (cross-corr)
      const float mag = sqrtf(gxv * gxv + gyv * gyv);
      const float ph  = (atan2f(gxv, gyv) / 3.14159265358979323846f) * 9.0f; // [-9, 9]
      const float flf = floorf(ph);
      const float cef = ceilf(ph);
      int fl = (int)flf % 9; fl = (fl + 9) % 9;   // Python-style non-negative mod
      int ce = (int)cef % 9; ce = (ce + 9) % 9;
#pragma unroll
      for (int o = 0; o < ORI; ++o) {
        float w = ((o == fl) ? mag : 0.0f) + ((o == ce) ? mag : 0.0f);
        acc[o] += w;
      }
    }

#pragma unroll
    for (int j = 0; j < 10; ++j) { r0[j] = r1[j]; r1[j] = r2[j]; }
  }

  // ---------------- write pooled cell histogram: out[n][o][cy][cx] ----------------
  const int cy = tcy + lcy;
  const int cx = tcx + lcx;
  float* op = out + (size_t)n * ORI * NCELL * NCELL + (size_t)cy * NCELL + cx;
#pragma unroll
  for (int o = 0; o < ORI; ++o)
    op[(size_t)o * NCELL * NCELL] = acc[o] * (1.0f / 64.0f);
}

extern "C" void kernel_launch(void* const* d_in, const int* in_sizes, int n_in,
                              void* d_out, int out_size, void* d_ws, size_t ws_size,
                              hipStream_t stream) {
  (void)in_sizes; (void)n_in; (void)d_ws; (void)ws_size; (void)out_size;
  const float* x = (const float*)d_in[0];   // (64,1,512,512) fp32; d_in[1] (Sobel) hardcoded
  float* out = (float*)d_out;               // (64, 9*64*64) fp32

  dim3 grid(NIMG * 16);                     // 64 images x 4x4 tiles
  dim3 block(256);                          // 8 wave32 waves; one 8x8 cell per thread
  size_t shmem = (size_t)LROWS * LSTRIDE * sizeof(float);  // ~68.6 KB dynamic LDS
  hog_fused_kernel<<<grid, block, shmem, stream>>>(x, out);
}